// MovieLensHetGNN_49752901157160
// MI455X (gfx1250) — compile-verified
//
#include <hip/hip_runtime.h>

// ---------------- problem constants (match reference) ----------------
#define N_USERS  100000
#define N_MOVIES 20000
#define D_USER   128
#define D_MOVIE  404
#define HID      256
#define OUT_D    128

// ---------------- workspace layout (float offsets) -------------------
#define OFF_CNT_MV  ((size_t)0)                         // 20000   (becomes inv count)
#define OFF_CNT_US  ((size_t)20000)                     // 100000  (becomes inv count)
#define OFF_AGG_MV  ((size_t)120000)                    // 20000*128  = 2,560,000
#define OFF_AGG_US  ((size_t)2680000)                   // 100000*256 = 25,600,000
#define OFF_YM      ((size_t)28280000)                  // 20000*256  = 5,120,000
#define OFF_HM      ((size_t)33400000)                  // 20000*256  = 5,120,000
#define OFF_HU      ((size_t)38520000)                  // 100000*256 = 25,600,000
#define OFF_WP      ((size_t)64120000)                  // packed weights (float2 region)
// layer-2 scratch reuses the (dead after layer 1) aggregation region:
#define OFF_TU      OFF_AGG_MV                          // 100000*128 = 12,800,000
#define OFF_TM      ((size_t)12920000)                  // 20000*128  = 2,560,000 (< OFF_YM)
#define ZERO_N      OFF_YM                              // zero counts + agg buffers

// packed-weight sub-offsets, in float2 units, within OFF_WP region
#define WP1_MU_L  ((size_t)0)        // 202*256 = 51712
#define WP1_RM_L  ((size_t)51712)    // 64*256  = 16384
#define WP1_RM_R  ((size_t)68096)    // 202*256 = 51712
#define WP1_MU_R  ((size_t)119808)   // 64*256  = 16384
#define WP2_RM_L  ((size_t)136192)   // 128*128 = 16384
#define WP2_MU_L  ((size_t)152576)
#define WP2_RM_R  ((size_t)168960)
#define WP2_MU_R  ((size_t)185344)   // end = 201728 f2 = 403456 floats

typedef float v2f __attribute__((ext_vector_type(2)));
typedef float v8f __attribute__((ext_vector_type(8)));

#define KC 32            // K-chunk staged in LDS (8 WMMA steps)
#define LDS_STRIDE 36    // padded row stride (floats): 36*m mod 64 -> 0,4,..,60 (conflict-free)

// ---------------------------------------------------------------------
// zero-fill (graph-capture safe replacement for hipMemsetAsync)
// ---------------------------------------------------------------------
__global__ __launch_bounds__(256) void zero_f32(float* __restrict__ p, size_t n) {
  size_t i = (size_t)blockIdx.x * 256 + threadIdx.x;
  if (i < n) p[i] = 0.0f;
}

// ---------------------------------------------------------------------
// repack W[K][N] row-major -> Wp[K/2][N] of float2 (K-pairs interleaved)
// so a WMMA B-fragment is one coalesced b64 load per lane.
// ---------------------------------------------------------------------
__global__ __launch_bounds__(256) void repack_w(const float* __restrict__ W,
                                                float2* __restrict__ Wp, int K, int N) {
  int i = blockIdx.x * 256 + threadIdx.x;
  int total = (K >> 1) * N;
  if (i < total) {
    int kp = i / N;
    int n  = i - kp * N;
    Wp[i] = make_float2(W[(size_t)(2 * kp) * N + n], W[(size_t)(2 * kp + 1) * N + n]);
  }
}

// ---------------------------------------------------------------------
// per-destination edge counts (float so we can reuse as inverse later)
// ---------------------------------------------------------------------
__global__ __launch_bounds__(256) void count_edges(const int* __restrict__ esrc,
                                                   const int* __restrict__ edst,
                                                   float* __restrict__ cnt_mv,
                                                   float* __restrict__ cnt_us, int E) {
  int e = blockIdx.x * 256 + threadIdx.x;
  if (e < E) {
    atomicAdd(&cnt_mv[edst[e]], 1.0f);
    atomicAdd(&cnt_us[esrc[e]], 1.0f);
  }
}

__global__ __launch_bounds__(256) void invert_counts(float* __restrict__ cnt, int n) {
  int i = blockIdx.x * 256 + threadIdx.x;
  if (i < n) cnt[i] = 1.0f / fmaxf(cnt[i], 1.0f);
}

// ---------------------------------------------------------------------
// WMMA f32 GEMM:  C = relu?( A0@W0 [+ A1@W1] [+ Cinit] [+ bias] )
//   A row-major [M,K]; W pre-packed as float2 [K/2][N]; C row-major [M,N].
//   block (32,8): 8 waves x 2 M-tiles cover a 32x128 C tile. A chunk
//   (32 x KC) staged in LDS once per block, shared by all waves; each B
//   fragment (one coalesced b64/lane) feeds TWO WMMAs. Full chunks run a
//   fully unrolled 8-step body so ~24 loads are in flight per WMMA burst.
//   Requires M%32==0, K%4==0; grid.y*128 == N.
// V_WMMA_F32_16X16X4_F32 fragment layout (ISA 7.12.2):
//   A 16x4 : lane&15 = M row; pair K = (lane>>4)*2 + {0,1}
//   B 4x16 : lane&15 = N col; pair K = (lane>>4)*2 + {0,1}
//   C 16x16: lane&15 = N col; VGPR r holds M = r + (lane>=16 ? 8 : 0)
// ---------------------------------------------------------------------
__global__ __launch_bounds__(256)
void gemm_wmma_f32(const float* __restrict__ A0, int K0, const float2* __restrict__ W0,
                   const float* __restrict__ A1, int K1, const float2* __restrict__ W1,
                   const float* __restrict__ Cinit, const float* __restrict__ bias,
                   float* __restrict__ C, int N, int relu) {
  __shared__ float As[32 * LDS_STRIDE];

  const int lane = threadIdx.x;                       // 0..31
  const int wave = threadIdx.y;                       // 0..7
  const int tid  = wave * 32 + lane;
  const int row0 = blockIdx.x * 32;
  const int col0 = (blockIdx.y * 8 + wave) * 16;

  const int am = lane & 15;                           // M row (A frag) / N col (B frag)
  const int kk = (lane >> 4) << 1;                    // 0 or 2 -> K sub-slot
  const int col = col0 + am;

  // cooperative A-stage mapping: 16 threads per row, each loads one float2,
  // each thread covers rows lm and lm+16.
  const int lm = tid >> 4;                            // 0..15
  const int lk = (tid & 15) << 1;                     // 0,2,..,30

  v8f c0 = {0.f, 0.f, 0.f, 0.f, 0.f, 0.f, 0.f, 0.f};
  v8f c1 = {0.f, 0.f, 0.f, 0.f, 0.f, 0.f, 0.f, 0.f};

  for (int pass = 0; pass < 2; ++pass) {
    const float* __restrict__ A = pass ? A1 : A0;
    if (A == nullptr) continue;                       // uniform branch
    const int K = pass ? K1 : K0;
    const float2* __restrict__ W = pass ? W1 : W0;
    const float* __restrict__ Arow0 = A + (size_t)(row0 + lm) * K;
    const float* __restrict__ Arow1 = A + (size_t)(row0 + 16 + lm) * K;

    for (int k0 = 0; k0 < K; k0 += KC) {
      const int kc_len = (K - k0) < KC ? (K - k0) : KC;   // multiple of 4
      __syncthreads();
      if (lk < kc_len) {
        *(float2*)&As[lm * LDS_STRIDE + lk]        = *(const float2*)(Arow0 + k0 + lk);
        *(float2*)&As[(lm + 16) * LDS_STRIDE + lk] = *(const float2*)(Arow1 + k0 + lk);
      }
      if (k0 + KC < K) {
        __builtin_prefetch(Arow0 + k0 + KC + lk, 0, 0);
        __builtin_prefetch(Arow1 + k0 + KC + lk, 0, 0);
      }
      __syncthreads();

      const float2* __restrict__ Wk = W + ((size_t)((k0 + kk) >> 1)) * N + col;
      if (kc_len == KC) {
        // ---- fast path: fully unrolled, loads issued as clauses ----
        float2 bw[8];
        v2f a0[8], a1[8];
#pragma unroll
        for (int i = 0; i < 8; ++i) bw[i] = Wk[(size_t)(2 * i) * N];
#pragma unroll
        for (int i = 0; i < 8; ++i) {
          a0[i] = *(const v2f*)&As[am * LDS_STRIDE + 4 * i + kk];
          a1[i] = *(const v2f*)&As[(am + 16) * LDS_STRIDE + 4 * i + kk];
        }
#pragma unroll
        for (int i = 0; i < 8; ++i) {
          v2f b;
          b.x = bw[i].x;
          b.y = bw[i].y;
          c0 = __builtin_amdgcn_wmma_f32_16x16x4_f32(false, a0[i], false, b, (short)0,
                                                     c0, false, false);
          c1 = __builtin_amdgcn_wmma_f32_16x16x4_f32(false, a1[i], false, b, (short)0,
                                                     c1, false, false);
        }
      } else {
        // ---- remainder path (K % 32, still a multiple of 4) ----
        for (int kc = 0; kc < kc_len; kc += 4) {
          v2f a0 = *(const v2f*)&As[am * LDS_STRIDE + kc + kk];
          v2f a1 = *(const v2f*)&As[(am + 16) * LDS_STRIDE + kc + kk];
          float2 bw = Wk[(size_t)(kc >> 1) * N];
          v2f b;
          b.x = bw.x;
          b.y = bw.y;
          c0 = __builtin_amdgcn_wmma_f32_16x16x4_f32(false, a0, false, b, (short)0,
                                                     c0, false, false);
          c1 = __builtin_amdgcn_wmma_f32_16x16x4_f32(false, a1, false, b, (short)0,
                                                     c1, false, false);
        }
      }
    }
  }

  const int cn = lane & 15;
  const int mb = (lane >> 4) << 3;                    // 0 or 8
  const float bval = bias ? bias[col0 + cn] : 0.0f;
#pragma unroll
  for (int r = 0; r < 8; ++r) {
    size_t idx0 = (size_t)(row0 + mb + r) * N + col0 + cn;
    size_t idx1 = idx0 + (size_t)16 * N;
    float v0 = c0[r] + bval;
    float v1 = c1[r] + bval;
    if (Cinit) {
      v0 += Cinit[idx0];
      v1 += Cinit[idx1];
    }
    if (relu) {
      v0 = fmaxf(v0, 0.0f);
      v1 = fmaxf(v1, 0.0f);
    }
    C[idx0] = v0;
    C[idx1] = v1;
  }
}

// ---------------------------------------------------------------------
// Layer-1 scatter: one wave32 per edge (mean folded in via inv counts).
//   agg_mv[dst, 0:128] += x_user[src, :] * inv_mv[dst]   (4 f32 / lane)
//   agg_us[src, 0:256] += ym[dst, :]     * inv_us[src]   (8 f32 / lane)
// ---------------------------------------------------------------------
__global__ __launch_bounds__(256)
void scatter_l1(const int* __restrict__ esrc, const int* __restrict__ edst,
                const float* __restrict__ xu, const float* __restrict__ ym,
                const float* __restrict__ inv_mv, const float* __restrict__ inv_us,
                float* __restrict__ agg_mv, float* __restrict__ agg_us, int E) {
  int e = (int)((blockIdx.x * 256 + threadIdx.x) >> 5);
  int lane = threadIdx.x & 31;
  if (e >= E) return;
  int s = esrc[e], d = edst[e];
  float wm = inv_mv[d];
  float wu = inv_us[s];

  float4 v = ((const float4*)(xu + (size_t)s * D_USER))[lane];
  float* om = agg_mv + (size_t)d * D_USER + lane * 4;
  atomicAdd(om + 0, v.x * wm);
  atomicAdd(om + 1, v.y * wm);
  atomicAdd(om + 2, v.z * wm);
  atomicAdd(om + 3, v.w * wm);

  const float4* yr = (const float4*)(ym + (size_t)d * HID);
  float4 u0 = yr[lane];
  float4 u1 = yr[lane + 32];
  float* ou = agg_us + (size_t)s * HID + lane * 4;
  atomicAdd(ou + 0, u0.x * wu);
  atomicAdd(ou + 1, u0.y * wu);
  atomicAdd(ou + 2, u0.z * wu);
  atomicAdd(ou + 3, u0.w * wu);
  float* ou2 = ou + 128;
  atomicAdd(ou2 + 0, u1.x * wu);
  atomicAdd(ou2 + 1, u1.y * wu);
  atomicAdd(ou2 + 2, u1.z * wu);
  atomicAdd(ou2 + 3, u1.w * wu);
}

// ---------------------------------------------------------------------
// Layer-2 scatter, accumulating straight into d_out (root term + bias
// already written there by the preceding GEMMs).
// ---------------------------------------------------------------------
__global__ __launch_bounds__(256)
void scatter_l2(const int* __restrict__ esrc, const int* __restrict__ edst,
                const float* __restrict__ tu, const float* __restrict__ tm,
                const float* __restrict__ inv_mv, const float* __restrict__ inv_us,
                float* __restrict__ out_mv, float* __restrict__ out_us, int E) {
  int e = (int)((blockIdx.x * 256 + threadIdx.x) >> 5);
  int lane = threadIdx.x & 31;
  if (e >= E) return;
  int s = esrc[e], d = edst[e];
  float wm = inv_mv[d];
  float wu = inv_us[s];

  float4 a = ((const float4*)(tu + (size_t)s * OUT_D))[lane];
  float* om = out_mv + (size_t)d * OUT_D + lane * 4;
  atomicAdd(om + 0, a.x * wm);
  atomicAdd(om + 1, a.y * wm);
  atomicAdd(om + 2, a.z * wm);
  atomicAdd(om + 3, a.w * wm);

  float4 b = ((const float4*)(tm + (size_t)d * OUT_D))[lane];
  float* ou = out_us + (size_t)s * OUT_D + lane * 4;
  atomicAdd(ou + 0, b.x * wu);
  atomicAdd(ou + 1, b.y * wu);
  atomicAdd(ou + 2, b.z * wu);
  atomicAdd(ou + 3, b.w * wu);
}

// ---------------------------------------------------------------------
extern "C" void kernel_launch(void* const* d_in, const int* in_sizes, int n_in,
                              void* d_out, int out_size, void* d_ws, size_t ws_size,
                              hipStream_t stream) {
  const float* x_user  = (const float*)d_in[0];
  const float* x_movie = (const float*)d_in[1];
  const int*   esrc    = (const int*)d_in[2];
  const int*   edst    = (const int*)d_in[3];
  const float* w1_rm_l = (const float*)d_in[4];
  const float* b1_rm   = (const float*)d_in[5];
  const float* w1_rm_r = (const float*)d_in[6];
  const float* w1_mu_l = (const float*)d_in[7];
  const float* b1_mu   = (const float*)d_in[8];
  const float* w1_mu_r = (const float*)d_in[9];
  const float* w2_rm_l = (const float*)d_in[10];
  const float* b2_rm   = (const float*)d_in[11];
  const float* w2_rm_r = (const float*)d_in[12];
  const float* w2_mu_l = (const float*)d_in[13];
  const float* b2_mu   = (const float*)d_in[14];
  const float* w2_mu_r = (const float*)d_in[15];

  const int E = in_sizes[2];

  float* ws     = (float*)d_ws;
  float* cnt_mv = ws + OFF_CNT_MV;   // later inv counts
  float* cnt_us = ws + OFF_CNT_US;
  float* agg_mv = ws + OFF_AGG_MV;
  float* agg_us = ws + OFF_AGG_US;
  float* ym     = ws + OFF_YM;
  float* h_mv   = ws + OFF_HM;
  float* h_us   = ws + OFF_HU;
  float* t_us   = ws + OFF_TU;       // aliases agg region (dead after L1)
  float* t_mv   = ws + OFF_TM;

  float2* wp      = (float2*)(ws + OFF_WP);
  float2* wp1mu_l = wp + WP1_MU_L;
  float2* wp1rm_l = wp + WP1_RM_L;
  float2* wp1rm_r = wp + WP1_RM_R;
  float2* wp1mu_r = wp + WP1_MU_R;
  float2* wp2rm_l = wp + WP2_RM_L;
  float2* wp2mu_l = wp + WP2_MU_L;
  float2* wp2rm_r = wp + WP2_RM_R;
  float2* wp2mu_r = wp + WP2_MU_R;

  float* out_mv = (float*)d_out;                             // [20000,128]
  float* out_us = (float*)d_out + (size_t)N_MOVIES * OUT_D;  // [100000,128]

  dim3 blk(256);
  dim3 gblk(32, 8);

  // 0) zero counts + layer-1 aggregation buffers
  zero_f32<<<dim3((unsigned)((ZERO_N + 255) / 256)), blk, 0, stream>>>(ws, ZERO_N);

  // 0b) repack all weight matrices into float2-interleaved fragment layout
  repack_w<<<dim3((D_MOVIE / 2 * HID + 255) / 256), blk, 0, stream>>>(w1_mu_l, wp1mu_l, D_MOVIE, HID);
  repack_w<<<dim3((D_USER / 2 * HID + 255) / 256), blk, 0, stream>>>(w1_rm_l, wp1rm_l, D_USER, HID);
  repack_w<<<dim3((D_MOVIE / 2 * HID + 255) / 256), blk, 0, stream>>>(w1_rm_r, wp1rm_r, D_MOVIE, HID);
  repack_w<<<dim3((D_USER / 2 * HID + 255) / 256), blk, 0, stream>>>(w1_mu_r, wp1mu_r, D_USER, HID);
  repack_w<<<dim3((HID / 2 * OUT_D + 255) / 256), blk, 0, stream>>>(w2_rm_l, wp2rm_l, HID, OUT_D);
  repack_w<<<dim3((HID / 2 * OUT_D + 255) / 256), blk, 0, stream>>>(w2_mu_l, wp2mu_l, HID, OUT_D);
  repack_w<<<dim3((HID / 2 * OUT_D + 255) / 256), blk, 0, stream>>>(w2_rm_r, wp2rm_r, HID, OUT_D);
  repack_w<<<dim3((HID / 2 * OUT_D + 255) / 256), blk, 0, stream>>>(w2_mu_r, wp2mu_r, HID, OUT_D);

  // 1) edge degree counts, then invert in place (contiguous 120000 floats)
  count_edges<<<dim3((E + 255) / 256), blk, 0, stream>>>(esrc, edst, cnt_mv, cnt_us, E);
  invert_counts<<<dim3((N_MOVIES + N_USERS + 255) / 256), blk, 0, stream>>>(
      cnt_mv, N_MOVIES + N_USERS);

  // 2) pre-transform movie features for user-side aggregation: ym = x_movie @ w1_mu_l
  gemm_wmma_f32<<<dim3(N_MOVIES / 32, HID / 128), gblk, 0, stream>>>(
      x_movie, D_MOVIE, wp1mu_l, nullptr, 0, nullptr, nullptr, nullptr, ym, HID, 0);

  // 3) layer-1 scatter (mean folded in via inverse counts)
  scatter_l1<<<dim3((E * 32 + 255) / 256), blk, 0, stream>>>(
      esrc, edst, x_user, ym, cnt_mv, cnt_us, agg_mv, agg_us, E);

  // 4) h_movie = relu(agg_mv @ w1_rm_l + x_movie @ w1_rm_r + b1_rm)  (fused dual-K GEMM)
  gemm_wmma_f32<<<dim3(N_MOVIES / 32, HID / 128), gblk, 0, stream>>>(
      agg_mv, D_USER, wp1rm_l, x_movie, D_MOVIE, wp1rm_r, nullptr, b1_rm, h_mv, HID, 1);

  // 5) h_user = relu(agg_us + x_user @ w1_mu_r + b1_mu)
  gemm_wmma_f32<<<dim3(N_USERS / 32, HID / 128), gblk, 0, stream>>>(
      x_user, D_USER, wp1mu_r, nullptr, 0, nullptr, agg_us, b1_mu, h_us, HID, 1);

  // 6) layer-2 pre-transforms (256 -> 128 before scatter)
  gemm_wmma_f32<<<dim3(N_USERS / 32, OUT_D / 128), gblk, 0, stream>>>(
      h_us, HID, wp2rm_l, nullptr, 0, nullptr, nullptr, nullptr, t_us, OUT_D, 0);
  gemm_wmma_f32<<<dim3(N_MOVIES / 32, OUT_D / 128), gblk, 0, stream>>>(
      h_mv, HID, wp2mu_l, nullptr, 0, nullptr, nullptr, nullptr, t_mv, OUT_D, 0);

  // 7) root terms + bias straight into d_out
  gemm_wmma_f32<<<dim3(N_MOVIES / 32, OUT_D / 128), gblk, 0, stream>>>(
      h_mv, HID, wp2rm_r, nullptr, 0, nullptr, nullptr, b2_rm, out_mv, OUT_D, 0);
  gemm_wmma_f32<<<dim3(N_USERS / 32, OUT_D / 128), gblk, 0, stream>>>(
      h_us, HID, wp2mu_r, nullptr, 0, nullptr, nullptr, b2_mu, out_us, OUT_D, 0);

  // 8) layer-2 scatter-mean accumulated into d_out
  scatter_l2<<<dim3((E * 32 + 255) / 256), blk, 0, stream>>>(
      esrc, edst, t_us, t_mv, cnt_mv, cnt_us, out_mv, out_us, E);
}